// Qwen3_5MoeSparseMoeBlock_78752520339563
// MI455X (gfx1250) — compile-verified
//
#include <hip/hip_runtime.h>

// ---------------------------------------------------------------------------
// MoE block for MI455X (gfx1250, wave32, WMMA).
// fp32 GEMMs via V_WMMA_F32_16X16X4_F32. Problem is HBM-bound (1.35 GB of
// weights @ 23.3 TB/s); pipeline weight streaming with compute using the
// CDNA5 async global->LDS engine (ASYNCcnt) when available, register-staged
// double buffering otherwise.
// ---------------------------------------------------------------------------

#define T_TOK 4096
#define DMODEL 2048
#define NEXP 64
#define TOPK 8
#define FEXP 768
#define FSH 5632
#define CAP 1024

typedef float v2f __attribute__((ext_vector_type(2)));
typedef float v8f __attribute__((ext_vector_type(8)));
typedef __attribute__((__vector_size__(4 * sizeof(int)))) int v4i_t;

#if defined(__has_builtin)
#if __has_builtin(__builtin_amdgcn_global_load_async_to_lds_b128)
#define USE_ASYNC_LDS 1
#endif
#endif
#ifndef USE_ASYNC_LDS
#define USE_ASYNC_LDS 0
#endif

#if USE_ASYNC_LDS
#define NBUF 2
#else
#define NBUF 1
#endif

__device__ __forceinline__ void async_copy_b128(const float* src, float* dst) {
#if USE_ASYNC_LDS
  __builtin_amdgcn_global_load_async_to_lds_b128(
      (__attribute__((address_space(1))) v4i_t*)src,
      (__attribute__((address_space(3))) v4i_t*)dst, 0, 0);
#else
  (void)src;
  (void)dst;
#endif
}

__device__ __forceinline__ void async_wait_all() {
#if USE_ASYNC_LDS
  asm volatile("s_wait_asynccnt 0x0" ::: "memory");
#endif
}

__device__ __forceinline__ v8f wmma_f32(v2f a, v2f b, v8f c) {
  return __builtin_amdgcn_wmma_f32_16x16x4_f32(false, a, false, b, (short)0, c,
                                               false, false);
}

// ---------------------------------------------------------------------------
// Kernel 0: zero expert counters
// ---------------------------------------------------------------------------
__global__ void init_counts_kernel(int* __restrict__ counts) {
  counts[threadIdx.x] = 0;
}

// ---------------------------------------------------------------------------
// Kernel 1: routing — gate logits, softmax, top-8, normalized weights,
// capacity assignment (atomic; CAP=1024 vs mean 512/expert => no drops),
// plus shared-expert sigmoid gate. One block (64 threads) per token.
// ---------------------------------------------------------------------------
__global__ void routing_kernel(const float* __restrict__ x,
                               const float* __restrict__ gate_w,
                               const float* __restrict__ sgw,
                               int* __restrict__ counts,
                               int* __restrict__ expert_rows,
                               int* __restrict__ pos_flat,
                               int* __restrict__ topk_idx,
                               float* __restrict__ topk_w,
                               float* __restrict__ sgate) {
  __shared__ float xr[DMODEL];
  __shared__ float sc[NEXP];
  __shared__ float red[NEXP];
  const int t = blockIdx.x;
  const int tid = threadIdx.x;
  const float* xrow = x + (long)t * DMODEL;
#pragma unroll
  for (int i = 0; i < DMODEL / 256; ++i) {
    const int idx = tid * 4 + i * 256;
    *(float4*)&xr[idx] = *(const float4*)&xrow[idx];
  }
  __syncthreads();
  float acc = 0.f;
  for (int d = 0; d < DMODEL; ++d) acc += xr[d] * gate_w[d * NEXP + tid];
  sc[tid] = acc;
  float sp = 0.f;
  for (int d = tid; d < DMODEL; d += 64) sp += xr[d] * sgw[d];
  red[tid] = sp;
  __syncthreads();
  if (tid == 0) {
    float ss = 0.f;
    for (int i = 0; i < 64; ++i) ss += red[i];
    sgate[t] = 1.f / (1.f + __expf(-ss));
    float mx = sc[0];
    for (int i = 1; i < NEXP; ++i) mx = fmaxf(mx, sc[i]);
    float p[NEXP];
    for (int i = 0; i < NEXP; ++i) p[i] = __expf(sc[i] - mx);
    // softmax denom cancels under top-k renormalization
    float wsum = 0.f;
    int sel[TOPK];
    float wv[TOPK];
    for (int k = 0; k < TOPK; ++k) {
      int best = 0;
      float bv = -1.f;
      for (int i = 0; i < NEXP; ++i)
        if (p[i] > bv) { bv = p[i]; best = i; }
      sel[k] = best;
      wv[k] = bv;
      wsum += bv;
      p[best] = -1.f;
    }
    for (int k = 0; k < TOPK; ++k) {
      const int fi = t * TOPK + k;
      topk_idx[fi] = sel[k];
      topk_w[fi] = wv[k] / wsum;
      const int pos = atomicAdd(&counts[sel[k]], 1);
      if (pos < CAP) {
        expert_rows[sel[k] * CAP + pos] = fi;
        pos_flat[fi] = pos;
      } else {
        pos_flat[fi] = -1;
      }
    }
  }
}

// ---------------------------------------------------------------------------
// Kernel 2: gated GEMM  h = silu(A@Bg) * (A@Bu)
//   A: M x 2048 (gathered token rows, or x directly when rows==nullptr)
//   B: 2048 x (2*Fhalf), g = cols [0,Fhalf), u = cols [Fhalf, 2*Fhalf)
// Block tile 128(M) x 64(N), K staged 32. Software pipeline: A reg-staged,
// B async-to-LDS double buffered (ASYNCcnt) or reg-staged fallback.
// 8 waves, each 32x32 per half via 2x2 grids of 16x16x4 fp32 WMMA.
// ---------------------------------------------------------------------------
__global__ __launch_bounds__(256) void gemm_gated_kernel(
    const float* __restrict__ X, const int* __restrict__ rows,
    const int* __restrict__ counts, int Mfull, const float* __restrict__ B,
    long bStride, int ldb, int Fhalf, float* __restrict__ H, long hStride,
    int ldh) {
  const int e = blockIdx.z;
  int M = Mfull;
  if (counts) { const int c = counts[e]; M = c < Mfull ? c : Mfull; }
  const int m0 = blockIdx.y * 128;
  if (m0 >= M) return;
  const int n0 = blockIdx.x * 64;
  const float* Be = B + (long)e * bStride;
  const int* rowsE = rows ? rows + e * CAP : nullptr;

  __shared__ float As[128][36];           // even stride: b64 frags, no conflicts
  __shared__ float Bg[NBUF][32][72];
  __shared__ float Bu[NBUF][32][72];

  const int tid = threadIdx.x;
  const int wave = tid >> 5, lane = tid & 31;
  const int wm = (wave >> 1) * 32, wn = (wave & 1) * 32;
  const int lm = lane & 15;
  const int kb = (lane >> 4) * 2;     // A/B K split per lane half
  const int rofs = (lane >> 4) << 3;  // C/D M split per lane half

  v8f acc_g[2][2], acc_u[2][2];
  const v8f vz = {0.f, 0.f, 0.f, 0.f, 0.f, 0.f, 0.f, 0.f};
#pragma unroll
  for (int i = 0; i < 2; ++i)
#pragma unroll
    for (int j = 0; j < 2; ++j) { acc_g[i][j] = vz; acc_u[i][j] = vz; }

  // A: 128 rows x 32 cols per stage; per-thread 4 x float4. Clamp OOB rows to
  // token row 0 (their outputs are masked in the epilogue).
  const int rA = tid >> 3;        // 0..31
  const int cA = (tid & 7) * 4;   // 0..28
  const float* aPtr[4];
#pragma unroll
  for (int p = 0; p < 4; ++p) {
    const int gm = m0 + rA + p * 32;
    long srcRow = 0;
    if (gm < M) srcRow = rowsE ? (long)(rowsE[gm] >> 3) * DMODEL : (long)gm * DMODEL;
    aPtr[p] = X + srcRow;
  }
  const int rB = tid >> 4;        // 0..15
  const int cB = (tid & 15) * 4;  // 0..60

  float4 aReg[4];
#if !USE_ASYNC_LDS
  float4 gReg[2], uReg[2];
#endif

  auto loadA = [&](int k0) {
#pragma unroll
    for (int p = 0; p < 4; ++p) aReg[p] = *(const float4*)(aPtr[p] + k0 + cA);
  };
  auto issueB = [&](int k0, int buf) {
#pragma unroll
    for (int p = 0; p < 2; ++p) {
      const int r = rB + p * 16;
      const float* brow = Be + (long)(k0 + r) * ldb + n0 + cB;
#if USE_ASYNC_LDS
      async_copy_b128(brow, &Bg[buf][r][cB]);
      async_copy_b128(brow + Fhalf, &Bu[buf][r][cB]);
#else
      gReg[p] = *(const float4*)brow;
      uReg[p] = *(const float4*)(brow + Fhalf);
#endif
    }
    (void)buf;
  };
  auto stageStore = [&]() {
#pragma unroll
    for (int p = 0; p < 4; ++p) {
      const int r = rA + p * 32;
      const float4 v = aReg[p];
      As[r][cA] = v.x; As[r][cA + 1] = v.y; As[r][cA + 2] = v.z; As[r][cA + 3] = v.w;
    }
#if !USE_ASYNC_LDS
#pragma unroll
    for (int p = 0; p < 2; ++p) {
      const int r = rB + p * 16;
      Bg[0][r][cB] = gReg[p].x; Bg[0][r][cB + 1] = gReg[p].y;
      Bg[0][r][cB + 2] = gReg[p].z; Bg[0][r][cB + 3] = gReg[p].w;
      Bu[0][r][cB] = uReg[p].x; Bu[0][r][cB + 1] = uReg[p].y;
      Bu[0][r][cB + 2] = uReg[p].z; Bu[0][r][cB + 3] = uReg[p].w;
    }
#endif
  };

  const int nStages = DMODEL / 32;
  loadA(0);
  issueB(0, 0);
  for (int s = 0; s < nStages; ++s) {
    const int cur = USE_ASYNC_LDS ? (s & 1) : 0;
    async_wait_all();     // stage-s async B tiles landed (own wave)
    __syncthreads();      // all waves: tiles landed, previous compute retired
    stageStore();         // A (and fallback B) regs -> LDS
    __syncthreads();
    if (s + 1 < nStages) {
      loadA((s + 1) * 32);
      issueB((s + 1) * 32, USE_ASYNC_LDS ? ((s + 1) & 1) : 0);
    }
#pragma unroll
    for (int kk = 0; kk < 32; kk += 4) {
      const v2f a0 = *(const v2f*)&As[wm + lm][kk + kb];
      const v2f a1 = *(const v2f*)&As[wm + 16 + lm][kk + kb];
      const v2f g0 = {Bg[cur][kk + kb][wn + lm], Bg[cur][kk + kb + 1][wn + lm]};
      const v2f g1 = {Bg[cur][kk + kb][wn + 16 + lm], Bg[cur][kk + kb + 1][wn + 16 + lm]};
      const v2f u0 = {Bu[cur][kk + kb][wn + lm], Bu[cur][kk + kb + 1][wn + lm]};
      const v2f u1 = {Bu[cur][kk + kb][wn + 16 + lm], Bu[cur][kk + kb + 1][wn + 16 + lm]};
      acc_g[0][0] = wmma_f32(a0, g0, acc_g[0][0]);
      acc_g[0][1] = wmma_f32(a0, g1, acc_g[0][1]);
      acc_g[1][0] = wmma_f32(a1, g0, acc_g[1][0]);
      acc_g[1][1] = wmma_f32(a1, g1, acc_g[1][1]);
      acc_u[0][0] = wmma_f32(a0, u0, acc_u[0][0]);
      acc_u[0][1] = wmma_f32(a0, u1, acc_u[0][1]);
      acc_u[1][0] = wmma_f32(a1, u0, acc_u[1][0]);
      acc_u[1][1] = wmma_f32(a1, u1, acc_u[1][1]);
    }
  }
  // epilogue: h = silu(g) * u
#pragma unroll
  for (int mi = 0; mi < 2; ++mi)
#pragma unroll
    for (int ni = 0; ni < 2; ++ni) {
      const int col = n0 + wn + ni * 16 + lm;
#pragma unroll
      for (int i = 0; i < 8; ++i) {
        const int row = m0 + wm + mi * 16 + rofs + i;
        if (row < M) {
          const float gv = acc_g[mi][ni][i];
          const float hv = gv / (1.f + __expf(-gv)) * acc_u[mi][ni][i];
          H[(long)e * hStride + (long)row * ldh + col] = hv;
        }
      }
    }
}

// ---------------------------------------------------------------------------
// Kernel 3: down GEMM  OUT = A @ B  (A = h, K = Kdim, N = 2048)
//   rows != nullptr : scatter row r -> y_pair[expert_rows[r]] (unscaled)
//   rows == nullptr : OUT[r] = row * rowScale[r]  (shared expert, sigmoid gate)
// ---------------------------------------------------------------------------
__global__ __launch_bounds__(256) void gemm_down_kernel(
    const float* __restrict__ H, long hStride, int ldh, int Kdim,
    const float* __restrict__ B, long bStride,
    const float* __restrict__ rowScale, const int* __restrict__ rows,
    const int* __restrict__ counts, int Mfull, float* __restrict__ OUT) {
  const int e = blockIdx.z;
  int M = Mfull;
  if (counts) { const int c = counts[e]; M = c < Mfull ? c : Mfull; }
  const int m0 = blockIdx.y * 128;
  if (m0 >= M) return;
  const int n0 = blockIdx.x * 64;
  const float* Be = B + (long)e * bStride;
  const int* rowsE = rows ? rows + e * CAP : nullptr;
  const float* He = H + (long)e * hStride;

  __shared__ float As[128][36];
  __shared__ float Bs[NBUF][32][72];

  const int tid = threadIdx.x;
  const int wave = tid >> 5, lane = tid & 31;
  const int wm = (wave >> 1) * 32, wn = (wave & 1) * 32;
  const int lm = lane & 15;
  const int kb = (lane >> 4) * 2;
  const int rofs = (lane >> 4) << 3;

  v8f acc[2][2];
  const v8f vz = {0.f, 0.f, 0.f, 0.f, 0.f, 0.f, 0.f, 0.f};
#pragma unroll
  for (int i = 0; i < 2; ++i)
#pragma unroll
    for (int j = 0; j < 2; ++j) acc[i][j] = vz;

  const int rA = tid >> 3;
  const int cA = (tid & 7) * 4;
  const int rB = tid >> 4;
  const int cB = (tid & 15) * 4;

  float4 aReg[4];
#if !USE_ASYNC_LDS
  float4 bReg[2];
#endif

  auto loadA = [&](int k0) {
#pragma unroll
    for (int p = 0; p < 4; ++p)
      aReg[p] = *(const float4*)(He + (long)(m0 + rA + p * 32) * ldh + k0 + cA);
  };
  auto issueB = [&](int k0, int buf) {
#pragma unroll
    for (int p = 0; p < 2; ++p) {
      const int r = rB + p * 16;
      const float* brow = Be + (long)(k0 + r) * DMODEL + n0 + cB;
#if USE_ASYNC_LDS
      async_copy_b128(brow, &Bs[buf][r][cB]);
#else
      bReg[p] = *(const float4*)brow;
#endif
    }
    (void)buf;
  };
  auto stageStore = [&]() {
#pragma unroll
    for (int p = 0; p < 4; ++p) {
      const int r = rA + p * 32;
      const float4 v = aReg[p];
      As[r][cA] = v.x; As[r][cA + 1] = v.y; As[r][cA + 2] = v.z; As[r][cA + 3] = v.w;
    }
#if !USE_ASYNC_LDS
#pragma unroll
    for (int p = 0; p < 2; ++p) {
      const int r = rB + p * 16;
      Bs[0][r][cB] = bReg[p].x; Bs[0][r][cB + 1] = bReg[p].y;
      Bs[0][r][cB + 2] = bReg[p].z; Bs[0][r][cB + 3] = bReg[p].w;
    }
#endif
  };

  const int nStages = Kdim / 32;
  loadA(0);
  issueB(0, 0);
  for (int s = 0; s < nStages; ++s) {
    const int cur = USE_ASYNC_LDS ? (s & 1) : 0;
    async_wait_all();
    __syncthreads();
    stageStore();
    __syncthreads();
    if (s + 1 < nStages) {
      loadA((s + 1) * 32);
      issueB((s + 1) * 32, USE_ASYNC_LDS ? ((s + 1) & 1) : 0);
    }
#pragma unroll
    for (int kk = 0; kk < 32; kk += 4) {
      const v2f a0 = *(const v2f*)&As[wm + lm][kk + kb];
      const v2f a1 = *(const v2f*)&As[wm + 16 + lm][kk + kb];
      const v2f b0 = {Bs[cur][kk + kb][wn + lm], Bs[cur][kk + kb + 1][wn + lm]};
      const v2f b1 = {Bs[cur][kk + kb][wn + 16 + lm], Bs[cur][kk + kb + 1][wn + 16 + lm]};
      acc[0][0] = wmma_f32(a0, b0, acc[0][0]);
      acc[0][1] = wmma_f32(a0, b1, acc[0][1]);
      acc[1][0] = wmma_f32(a1, b0, acc[1][0]);
      acc[1][1] = wmma_f32(a1, b1, acc[1][1]);
    }
  }
#pragma unroll
  for (int mi = 0; mi < 2; ++mi)
#pragma unroll
    for (int ni = 0; ni < 2; ++ni) {
      const int col = n0 + wn + ni * 16 + lm;
#pragma unroll
      for (int i = 0; i < 8; ++i) {
        const int row = m0 + wm + mi * 16 + rofs + i;
        if (row < M) {
          const float v = acc[mi][ni][i];
          if (rowsE) {
            OUT[(long)rowsE[row] * DMODEL + col] = v;
          } else {
            OUT[(long)row * DMODEL + col] = v * rowScale[row];
          }
        }
      }
    }
}

// ---------------------------------------------------------------------------
// Kernel 4: combine  y[t] = sum_k w[t,k]*y_pair[t*K+k] + shared_out[t]
// ---------------------------------------------------------------------------
__global__ void combine_kernel(const float* __restrict__ y_pair,
                               const float* __restrict__ topk_w,
                               const int* __restrict__ pos_flat,
                               const float* __restrict__ shared_out,
                               float* __restrict__ out) {
  const int t = blockIdx.x;
  const int tid = threadIdx.x;
  for (int d0 = tid * 4; d0 < DMODEL; d0 += 1024) {
    float4 a = *(const float4*)&shared_out[(long)t * DMODEL + d0];
#pragma unroll
    for (int k = 0; k < TOPK; ++k) {
      const int fi = t * TOPK + k;
      if (pos_flat[fi] >= 0) {
        const float w = topk_w[fi];
        const float4 v = *(const float4*)&y_pair[(long)fi * DMODEL + d0];
        a.x += w * v.x; a.y += w * v.y; a.z += w * v.z; a.w += w * v.w;
      }
    }
    *(float4*)&out[(long)t * DMODEL + d0] = a;
  }
}

// ---------------------------------------------------------------------------
// Host launcher
// ---------------------------------------------------------------------------
extern "C" void kernel_launch(void* const* d_in, const int* in_sizes, int n_in,
                              void* d_out, int out_size, void* d_ws,
                              size_t ws_size, hipStream_t stream) {
  const float* x = (const float*)d_in[0];
  const float* gate_w = (const float*)d_in[3];
  const float* w_gate_up = (const float*)d_in[4];
  const float* w_down = (const float*)d_in[5];
  const float* shared_gate_up = (const float*)d_in[6];
  const float* shared_down = (const float*)d_in[7];
  const float* shared_gate_w = (const float*)d_in[8];
  float* out = (float*)d_out;

  char* w = (char*)d_ws;
  int* counts = (int*)w;        w += 256;
  int* topk_idx = (int*)w;      w += (size_t)T_TOK * TOPK * 4;
  float* topk_w = (float*)w;    w += (size_t)T_TOK * TOPK * 4;
  int* pos_flat = (int*)w;      w += (size_t)T_TOK * TOPK * 4;
  int* expert_rows = (int*)w;   w += (size_t)NEXP * CAP * 4;
  float* sgate = (float*)w;     w += (size_t)T_TOK * 4;
  float* hbuf = (float*)w;      w += (size_t)NEXP * CAP * FEXP * 4;  // reused by shared FFN
  float* y_pair = (float*)w;    w += (size_t)T_TOK * TOPK * DMODEL * 4;
  float* shared_out = (float*)w;

  init_counts_kernel<<<1, 64, 0, stream>>>(counts);
  routing_kernel<<<T_TOK, 64, 0, stream>>>(x, gate_w, shared_gate_w, counts,
                                           expert_rows, pos_flat, topk_idx,
                                           topk_w, sgate);
  // expert gate_up + silu : h (E, CAP, F)
  gemm_gated_kernel<<<dim3(FEXP / 64, CAP / 128, NEXP), 256, 0, stream>>>(
      x, expert_rows, counts, CAP, w_gate_up, (long)DMODEL * 2 * FEXP,
      2 * FEXP, FEXP, hbuf, (long)CAP * FEXP, FEXP);
  // expert down : y_pair (T*K, D), scattered by assignment
  gemm_down_kernel<<<dim3(DMODEL / 64, CAP / 128, NEXP), 256, 0, stream>>>(
      hbuf, (long)CAP * FEXP, FEXP, FEXP, w_down, (long)FEXP * DMODEL, nullptr,
      expert_rows, counts, CAP, y_pair);
  // shared gate_up + silu : reuse hbuf as (T, FS)
  gemm_gated_kernel<<<dim3(FSH / 64, T_TOK / 128, 1), 256, 0, stream>>>(
      x, nullptr, nullptr, T_TOK, shared_gate_up, 0, 2 * FSH, FSH, hbuf, 0,
      FSH);
  // shared down, scaled by sigmoid gate : shared_out (T, D)
  gemm_down_kernel<<<dim3(DMODEL / 64, T_TOK / 128, 1), 256, 0, stream>>>(
      hbuf, 0, FSH, FSH, shared_down, 0, sgate, nullptr, nullptr, T_TOK,
      shared_out);
  combine_kernel<<<T_TOK, 256, 0, stream>>>(y_pair, topk_w, pos_flat,
                                            shared_out, out);
}